// VGAE_2465311228054
// MI455X (gfx1250) — compile-verified
//
#include <hip/hip_runtime.h>
#include <hip/hip_bf16.h>

// ---------------------------------------------------------------------------
// VGAE on MI455X (gfx1250, wave32). All dense GEMMs use V_WMMA_F32_16X16X4_F32
// (fp32 in/out: matches reference numerics; problem is L2/HBM bound, so fp32
// WMMA is the right precision — W_fc's 131MB stream and ~100M L2 atomics from
// the 1M-edge scatter dominate, not FLOPs).
// ---------------------------------------------------------------------------

typedef __attribute__((ext_vector_type(2))) float v2f;
typedef __attribute__((ext_vector_type(8))) float v8f;

#define NTOT   64000
#define NB     64
#define HDIM   64
#define FLAT   64000   // N_PER * H
#define FCDIM  512
#define LAT    64
#define ODIM   32

// ----- wave-level 16x16 f32 WMMA tile accumulation over K --------------------
// A: row-major [16 x K] slice starting at `A` with leading dim lda
// B: row-major [K x 16] slice starting at `B` with leading dim ldb
// A VGPR layout (16x4 f32): lanes 0-15 -> K={k,k+1}, lanes 16-31 -> K={k+2,k+3}
// B VGPR layout (4x16 f32): mirrored (N = lane&15)
__device__ __forceinline__ v8f wmma_accum(const float* __restrict__ A, int lda,
                                          const float* __restrict__ B, int ldb,
                                          int K, v8f acc) {
    const int lane  = threadIdx.x & 31;
    const int idx   = lane & 15;          // M for A-load, N for B-load
    const int half2 = (lane >> 4) << 1;   // 0 or 2 : K sub-offset
    const float* ap = A + (size_t)idx * lda + half2;
    const float* bp = B + (size_t)half2 * ldb + idx;
#pragma unroll 4
    for (int k = 0; k < K; k += 4) {
        v2f a, b;
        a.x = ap[k];
        a.y = ap[k + 1];
        b.x = bp[(size_t)k * ldb];
        b.y = bp[(size_t)(k + 1) * ldb];
        acc = __builtin_amdgcn_wmma_f32_16x16x4_f32(
            /*neg_a=*/false, a, /*neg_b=*/false, b,
            /*c_mod=*/(short)0, acc, /*reuse_a=*/false, /*reuse_b=*/false);
    }
    return acc;
}

// ----- generic GEMM: D = act(A @ B + bias), one wave per 16x16 tile ----------
__global__ __launch_bounds__(256) void gemm_wmma(
    const float* __restrict__ A, int lda,
    const float* __restrict__ B, int ldb,
    const float* __restrict__ bias,
    float* __restrict__ D, int ldd,
    int Mtiles, int Ntiles, int K, int do_relu) {
    const int wave = (blockIdx.x * blockDim.x + threadIdx.x) >> 5;
    const int mt = wave % Mtiles;
    const int nt = wave / Mtiles;
    if (nt >= Ntiles) return;                 // wave-uniform: EXEC stays all-1s
    v8f c = {};
    c = wmma_accum(A + (size_t)mt * 16 * lda, lda, B + nt * 16, ldb, K, c);
    const int lane = threadIdx.x & 31;
    const int n  = lane & 15;
    const int mb = (lane >> 4) * 8;           // C/D layout: lanes16-31 -> M+8
    const float bv = bias ? bias[nt * 16 + n] : 0.0f;
    float* dp = D + (size_t)(mt * 16 + mb) * ldd + nt * 16 + n;
#pragma unroll
    for (int g = 0; g < 8; ++g) {
        float v = c[g] + bv;
        if (do_relu) v = fmaxf(v, 0.0f);
        dp[(size_t)g * ldd] = v;
    }
}

// ----- big fc GEMM: X[64,64000] @ W_fc[64000,512], split-K=125 x 512 ---------
__global__ __launch_bounds__(256) void gemm_fc_splitk(
    const float* __restrict__ X, const float* __restrict__ W,
    float* __restrict__ acc) {
    const int wave = (blockIdx.x * blockDim.x + threadIdx.x) >> 5;
    const int mt = wave & 3;          // 4 M tiles
    const int nt = (wave >> 2) & 31;  // 32 N tiles
    const int ks = wave >> 7;         // 125 K slabs of 512
    if (ks >= 125) return;
    const float* A = X + (size_t)mt * 16 * FLAT + (size_t)ks * 512;
    const float* B = W + (size_t)ks * 512 * FCDIM + nt * 16;
    v8f c = {};
    c = wmma_accum(A, FLAT, B, FCDIM, 512, c);
    const int lane = threadIdx.x & 31;
    const int n  = lane & 15;
    const int mb = (lane >> 4) * 8;
    float* dst = acc + (size_t)(mt * 16 + mb) * FCDIM + nt * 16 + n;
#pragma unroll
    for (int g = 0; g < 8; ++g) atomicAdd(dst + (size_t)g * FCDIM, c[g]);
}

// ----- small elementwise kernels --------------------------------------------
__global__ void fill_val(float* p, float v, int n) {
    int i = blockIdx.x * blockDim.x + threadIdx.x;
    if (i < n) p[i] = v;
}

__global__ void degree_kernel(const int* __restrict__ s, const int* __restrict__ r,
                              float* deg_s, float* deg_r, int n_edges) {
    int i = blockIdx.x * blockDim.x + threadIdx.x;
    if (i < n_edges) {
        atomicAdd(&deg_s[s[i]], 1.0f);
        atomicAdd(&deg_r[r[i]], 1.0f);
    }
}

// encoder scatter: one wave per edge, 2 floats per lane (H=64)
__global__ __launch_bounds__(256) void scatter_enc(
    const float* __restrict__ h, const int* __restrict__ s,
    const int* __restrict__ r, const float* __restrict__ deg_s,
    float* __restrict__ hagg, int n_edges) {
    const int wave = (blockIdx.x * blockDim.x + threadIdx.x) >> 5;
    if (wave >= n_edges) return;
    const int lane = threadIdx.x & 31;
    const int si = s[wave], ri = r[wave];
    const float sc = rsqrtf(fmaxf(deg_s[si], 1.0f));
    const float* hp = h + (size_t)si * HDIM + lane * 2;
    float* op = hagg + (size_t)ri * HDIM + lane * 2;
    atomicAdd(op,     hp[0] * sc);
    atomicAdd(op + 1, hp[1] * sc);
}

// fold in self-edge and apply receiver normalization
__global__ void finalize_enc(float* __restrict__ hagg, const float* __restrict__ h,
                             const float* __restrict__ deg_s,
                             const float* __restrict__ deg_r, int n) {
    int i = blockIdx.x * blockDim.x + threadIdx.x;
    if (i < n) {
        int node = i >> 6;  // /HDIM
        float rs = rsqrtf(fmaxf(deg_s[node], 1.0f));
        float rr = rsqrtf(fmaxf(deg_r[node], 1.0f));
        hagg[i] = (hagg[i] + h[i] * rs) * rr;
    }
}

__global__ void finalize_fc(const float* __restrict__ acc,
                            const float* __restrict__ bias,
                            float* __restrict__ x, int n) {
    int i = blockIdx.x * blockDim.x + threadIdx.x;
    if (i < n) x[i] = fmaxf(acc[i] + bias[i & (FCDIM - 1)], 0.0f);
}

__global__ void reparam_kernel(const float* __restrict__ mean,
                               const float* __restrict__ logstd,
                               const float* __restrict__ eps,
                               float* __restrict__ z, int n) {
    int i = blockIdx.x * blockDim.x + threadIdx.x;
    if (i < n) z[i] = mean[i] + __expf(logstd[i]) * eps[i];
}

// decoder scatter: one wave per edge, 1 float per lane (ODIM=32), no self-edges
__global__ __launch_bounds__(256) void scatter_dec(
    const float* __restrict__ dd, const int* __restrict__ s,
    const int* __restrict__ r, const float* __restrict__ deg_s,
    float* __restrict__ out, int n_edges) {
    const int wave = (blockIdx.x * blockDim.x + threadIdx.x) >> 5;
    if (wave >= n_edges) return;
    const int lane = threadIdx.x & 31;
    const int si = s[wave], ri = r[wave];
    const float sc = rsqrtf(fmaxf(deg_s[si] - 1.0f, 1.0f));  // deg w/o self-edge
    atomicAdd(&out[(size_t)ri * ODIM + lane], dd[(size_t)si * ODIM + lane] * sc);
}

__global__ void finalize_dec(float* __restrict__ out,
                             const float* __restrict__ deg_r, int n) {
    int i = blockIdx.x * blockDim.x + threadIdx.x;
    if (i < n) out[i] *= rsqrtf(fmaxf(deg_r[i >> 5] - 1.0f, 1.0f));
}

// ---------------------------------------------------------------------------
extern "C" void kernel_launch(void* const* d_in, const int* in_sizes, int n_in,
                              void* d_out, int out_size, void* d_ws, size_t ws_size,
                              hipStream_t stream) {
    const float* nodes     = (const float*)d_in[0];
    const int*   senders   = (const int*)  d_in[1];
    const int*   receivers = (const int*)  d_in[2];
    const float* eps       = (const float*)d_in[3];
    const float* W_gcn1    = (const float*)d_in[4];
    const float* b_gcn1    = (const float*)d_in[5];
    const float* W_fc      = (const float*)d_in[6];
    const float* b_fc      = (const float*)d_in[7];
    const float* W_mean    = (const float*)d_in[8];
    const float* b_mean    = (const float*)d_in[9];
    const float* W_logstd  = (const float*)d_in[10];
    const float* b_logstd  = (const float*)d_in[11];
    const float* W_dec_fc  = (const float*)d_in[12];
    const float* b_dec_fc  = (const float*)d_in[13];
    const float* W_dec_out = (const float*)d_in[14];
    const float* b_dec_out = (const float*)d_in[15];
    const int n_edges = in_sizes[1];

    float* out    = (float*)d_out;
    float* mean   = out;                 // [64,64]
    float* logstd = out + 4096;          // [64,64]
    float* onodes = out + 8192;          // [64000,32]

    // workspace layout (floats): ~58 MB
    float* ws    = (float*)d_ws;
    float* h     = ws;                   // [64000,64]
    float* hagg  = ws + 4096000;         // [64000,64]
    float* deg_s = ws + 8192000;         // [64000]  (self-edge included)
    float* deg_r = ws + 8256000;         // [64000]
    float* xacc  = ws + 8320000;         // [64,512]
    float* x     = ws + 8352768;         // [64,512]
    float* z     = ws + 8385536;         // [64,64]
    float* d     = ws + 8389632;         // [64,64000]
    float* dd    = ws + 12485632;        // [64000,32]

    // --- init accumulators ---
    hipMemsetAsync(hagg,   0, (size_t)4096000 * sizeof(float), stream);
    hipMemsetAsync(xacc,   0, (size_t)32768   * sizeof(float), stream);
    hipMemsetAsync(onodes, 0, (size_t)2048000 * sizeof(float), stream);
    fill_val<<<(128000 + 255) / 256, 256, 0, stream>>>(deg_s, 1.0f, 128000); // deg_s|deg_r

    // --- degrees (with self-edges; decoder uses deg-1) ---
    degree_kernel<<<(n_edges + 255) / 256, 256, 0, stream>>>(
        senders, receivers, deg_s, deg_r, n_edges);

    // --- encoder GCN update: h = relu(nodes @ W_gcn1 + b) : 4000x4 tiles ---
    gemm_wmma<<<2000, 256, 0, stream>>>(nodes, HDIM, W_gcn1, HDIM, b_gcn1,
                                        h, HDIM, 4000, 4, 64, 1);

    // --- encoder aggregation ---
    scatter_enc<<<(n_edges * 32 + 255) / 256, 256, 0, stream>>>(
        h, senders, receivers, deg_s, hagg, n_edges);
    finalize_enc<<<(4096000 + 255) / 256, 256, 0, stream>>>(
        hagg, h, deg_s, deg_r, 4096000);

    // --- x = relu(X @ W_fc + b_fc), split-K WMMA (16000 waves) ---
    gemm_fc_splitk<<<2000, 256, 0, stream>>>(hagg, W_fc, xacc);
    finalize_fc<<<(32768 + 255) / 256, 256, 0, stream>>>(xacc, b_fc, x, 32768);

    // --- heads: mean / log_std (4x4 tiles each, K=512) ---
    gemm_wmma<<<2, 256, 0, stream>>>(x, FCDIM, W_mean, LAT, b_mean,
                                     mean, LAT, 4, 4, FCDIM, 0);
    gemm_wmma<<<2, 256, 0, stream>>>(x, FCDIM, W_logstd, LAT, b_logstd,
                                     logstd, LAT, 4, 4, FCDIM, 0);

    // --- reparameterize ---
    reparam_kernel<<<(4096 + 255) / 256, 256, 0, stream>>>(mean, logstd, eps, z, 4096);

    // --- decoder fc: d = relu(z @ W_dec_fc + b) : 4x4000 tiles, K=64 ---
    gemm_wmma<<<2000, 256, 0, stream>>>(z, LAT, W_dec_fc, FLAT, b_dec_fc,
                                        d, FLAT, 4, 4000, 64, 1);

    // --- decoder out: dd = d.reshape(64000,64) @ W_dec_out + b : 4000x2, K=64 ---
    gemm_wmma<<<1000, 256, 0, stream>>>(d, HDIM, W_dec_out, ODIM, b_dec_out,
                                        dd, ODIM, 4000, 2, 64, 0);

    // --- decoder aggregation (no self-edges) ---
    scatter_dec<<<(n_edges * 32 + 255) / 256, 256, 0, stream>>>(
        dd, senders, receivers, deg_s, onodes, n_edges);
    finalize_dec<<<(2048000 + 255) / 256, 256, 0, stream>>>(onodes, deg_r, 2048000);
}